// STGCNBlock_21010980012575
// MI455X (gfx1250) — compile-verified
//
#include <hip/hip_runtime.h>
#include <hip/hip_bf16.h>

typedef __attribute__((ext_vector_type(2))) float v2f;
typedef __attribute__((ext_vector_type(8))) float v8f;

#define NN 16000
#define NCHUNK 16      // 16 chunks of 1000 n-values
#define RED_THREADS 256

// ---------------------------------------------------------------------------
// Kernel 1: v_partial[chunk*4+b][c] = sum_{n in chunk} adj[0,n] * x[b,0,n,c]
// 64 blocks (chunk*4 + b), 256 threads. Coalesced b128 loads of x (c is
// contiguous, 8 floats = 2x float4 per (b,n)).
// ---------------------------------------------------------------------------
__global__ __launch_bounds__(RED_THREADS)
void stgcn_reduce_kernel(const float* __restrict__ x,
                         const float* __restrict__ adj,
                         float* __restrict__ partial) {
    __shared__ float sm[8 * RED_THREADS];
    const int tid   = threadIdx.x;
    const int b     = blockIdx.x & 3;
    const int chunk = blockIdx.x >> 2;
    const int n0    = chunk * (NN / NCHUNK);
    const int n1    = n0 + (NN / NCHUNK);

    float acc[8];
#pragma unroll
    for (int c = 0; c < 8; ++c) acc[c] = 0.f;

    const float* xb = x + (size_t)b * NN * 8;
    for (int n = n0 + tid; n < n1; n += RED_THREADS) {
        float a = adj[n];  // row 0 of adj
        const float4* xp = (const float4*)(xb + (size_t)n * 8);
        float4 lo = xp[0];
        float4 hi = xp[1];
        acc[0] = fmaf(a, lo.x, acc[0]);
        acc[1] = fmaf(a, lo.y, acc[1]);
        acc[2] = fmaf(a, lo.z, acc[2]);
        acc[3] = fmaf(a, lo.w, acc[3]);
        acc[4] = fmaf(a, hi.x, acc[4]);
        acc[5] = fmaf(a, hi.y, acc[5]);
        acc[6] = fmaf(a, hi.z, acc[6]);
        acc[7] = fmaf(a, hi.w, acc[7]);
    }
#pragma unroll
    for (int c = 0; c < 8; ++c) sm[c * RED_THREADS + tid] = acc[c];
    __syncthreads();
    for (int s = RED_THREADS / 2; s > 0; s >>= 1) {
        if (tid < s) {
#pragma unroll
            for (int c = 0; c < 8; ++c)
                sm[c * RED_THREADS + tid] += sm[c * RED_THREADS + tid + s];
        }
        __syncthreads();
    }
    if (tid < 8) partial[blockIdx.x * 8 + tid] = sm[tid * RED_THREADS];
}

// ---------------------------------------------------------------------------
// Kernel 2: single wave (32 lanes, EXEC all ones). Three chained small GEMMs
// via V_WMMA_F32_16X16X4_F32, D->A relayout through LDS, then BN/ReLU epilogue
// and the [4,1,16]+[4,16,1] broadcast add into out[4,16,16].
// ---------------------------------------------------------------------------
__global__ __launch_bounds__(32)
void stgcn_epilogue_kernel(const float* __restrict__ partial,
                           const float* __restrict__ x,
                           const float* __restrict__ W,
                           const float* __restrict__ gcn_b,
                           const float* __restrict__ tcn_w,
                           const float* __restrict__ tcn_b,
                           const float* __restrict__ bn1_g,
                           const float* __restrict__ bn1_bt,
                           const float* __restrict__ bn1_m,
                           const float* __restrict__ bn1_v,
                           const float* __restrict__ res_w,
                           const float* __restrict__ res_b,
                           const float* __restrict__ bn2_g,
                           const float* __restrict__ bn2_bt,
                           const float* __restrict__ bn2_m,
                           const float* __restrict__ bn2_v,
                           float* __restrict__ out) {
    __shared__ float Vbuf[4][8];
    __shared__ float Gbuf[16][16];
    __shared__ float Ybuf[16][16];
    __shared__ float Rbuf[16][16];

    const int lane = threadIdx.x;

    // ---- accumulate block partials: v[b][c] -------------------------------
    {
        int b = lane >> 3;
        int c = lane & 7;
        float s = 0.f;
#pragma unroll
        for (int chunk = 0; chunk < NCHUNK; ++chunk)
            s += partial[(chunk * 4 + b) * 8 + c];
        Vbuf[b][c] = s;
    }
    __syncthreads();

    const int half  = lane >> 4;   // 0: K pair {k0,k0+1}, 1: {k0+2,k0+3}
    const int idx16 = lane & 15;   // A row / B column

    // ---- GEMM1: G(16x16) = Vpad(16x8) @ W(8x16) ---------------------------
    v8f c1 = {};
#pragma unroll
    for (int k0 = 0; k0 < 8; k0 += 4) {
        int ka = k0 + half * 2;
        v2f a, bm;
        a[0] = (idx16 < 4) ? Vbuf[idx16][ka]     : 0.f;
        a[1] = (idx16 < 4) ? Vbuf[idx16][ka + 1] : 0.f;
        bm[0] = W[ka * 16 + idx16];
        bm[1] = W[(ka + 1) * 16 + idx16];
        c1 = __builtin_amdgcn_wmma_f32_16x16x4_f32(false, a, false, bm,
                                                   (short)0, c1, false, false);
    }
    {   // D layout: lane<16 -> rows 0..7, lane>=16 -> rows 8..15
        int row0 = half * 8;
#pragma unroll
        for (int r = 0; r < 8; ++r) Gbuf[row0 + r][idx16] = c1[r];
    }
    __syncthreads();

    // ---- GEMM2: Y = (G + gcn_bias)(16x16) @ B2(16x16), B2[i][o]=tcn_w[o,i,4]
    v8f c2 = {};
#pragma unroll
    for (int k0 = 0; k0 < 16; k0 += 4) {
        int ka = k0 + half * 2;
        v2f a, bm;
        a[0] = Gbuf[idx16][ka]     + gcn_b[ka];
        a[1] = Gbuf[idx16][ka + 1] + gcn_b[ka + 1];
        bm[0] = tcn_w[idx16 * 144 + ka * 9 + 4];
        bm[1] = tcn_w[idx16 * 144 + (ka + 1) * 9 + 4];
        c2 = __builtin_amdgcn_wmma_f32_16x16x4_f32(false, a, false, bm,
                                                   (short)0, c2, false, false);
    }

    // ---- GEMM3: R = X0pad(16x8) @ B3(8x16), B3[c][o]=res_w[o,c,0] ---------
    v8f c3 = {};
#pragma unroll
    for (int k0 = 0; k0 < 8; k0 += 4) {
        int ka = k0 + half * 2;
        v2f a, bm;
        a[0] = (idx16 < 4) ? x[(size_t)idx16 * NN * 8 + ka]     : 0.f;
        a[1] = (idx16 < 4) ? x[(size_t)idx16 * NN * 8 + ka + 1] : 0.f;
        bm[0] = res_w[idx16 * 8 + ka];
        bm[1] = res_w[idx16 * 8 + ka + 1];
        c3 = __builtin_amdgcn_wmma_f32_16x16x4_f32(false, a, false, bm,
                                                   (short)0, c3, false, false);
    }
    {
        int row0 = half * 8;
#pragma unroll
        for (int r = 0; r < 8; ++r) {
            Ybuf[row0 + r][idx16] = c2[r];
            Rbuf[row0 + r][idx16] = c3[r];
        }
    }
    __syncthreads();

    // ---- epilogue: out[b,i,j] = relu(bn1(y+tcn_b))[b,j] + bn2(r+res_b)[b,i]
#pragma unroll
    for (int t = 0; t < 32; ++t) {
        int idx = t * 32 + lane;
        int b = idx >> 8;
        int i = (idx >> 4) & 15;
        int j = idx & 15;
        float yv = Ybuf[b][j] + tcn_b[j];
        float ys = bn1_g[j] * rsqrtf(bn1_v[j] + 1e-5f);
        float yb = fmaf(yv - bn1_m[j], ys, bn1_bt[j]);
        float yr = fmaxf(yb, 0.f);
        float rv = Rbuf[b][i] + res_b[i];
        float rs = bn2_g[i] * rsqrtf(bn2_v[i] + 1e-5f);
        float rb = fmaf(rv - bn2_m[i], rs, bn2_bt[i]);
        out[idx] = yr + rb;
    }
}

extern "C" void kernel_launch(void* const* d_in, const int* in_sizes, int n_in,
                              void* d_out, int out_size, void* d_ws, size_t ws_size,
                              hipStream_t stream) {
    const float* x        = (const float*)d_in[0];   // [4,1,16000,8]
    const float* adj      = (const float*)d_in[1];   // [16000,16000] (row 0 used)
    const float* W        = (const float*)d_in[2];   // [8,16]
    const float* gcn_b    = (const float*)d_in[3];   // [16]
    const float* tcn_w    = (const float*)d_in[4];   // [16,16,9]
    const float* tcn_b    = (const float*)d_in[5];   // [16]
    const float* bn1_g    = (const float*)d_in[6];
    const float* bn1_bt   = (const float*)d_in[7];
    const float* bn1_m    = (const float*)d_in[8];
    const float* bn1_v    = (const float*)d_in[9];
    const float* res_w    = (const float*)d_in[10];  // [16,8,1]
    const float* res_b    = (const float*)d_in[11];
    const float* bn2_g    = (const float*)d_in[12];
    const float* bn2_bt   = (const float*)d_in[13];
    const float* bn2_m    = (const float*)d_in[14];
    const float* bn2_v    = (const float*)d_in[15];
    float* out            = (float*)d_out;           // [4,16,16]
    float* partial        = (float*)d_ws;            // 64*8 floats = 2 KB

    stgcn_reduce_kernel<<<64, RED_THREADS, 0, stream>>>(x, adj, partial);
    stgcn_epilogue_kernel<<<1, 32, 0, stream>>>(partial, x, W, gcn_b,
                                                tcn_w, tcn_b,
                                                bn1_g, bn1_bt, bn1_m, bn1_v,
                                                res_w, res_b,
                                                bn2_g, bn2_bt, bn2_m, bn2_v,
                                                out);
    (void)in_sizes; (void)n_in; (void)out_size; (void)ws_size;
}